// ObservableCalculator_43439299232036
// MI455X (gfx1250) — compile-verified
//
#include <hip/hip_runtime.h>
#include <hip/hip_bf16.h>
#include <stdint.h>

// Problem constants (match reference): B=8, SPIN=2, n=2048, N_ATOMS=256.
#define NN      2048
#define NPLANE  16            // B*SPIN
#define NATOMS  256
#define TM      16            // rows per workgroup tile
#define NC      256           // n-columns per workgroup tile
#define PITCH   17            // LDS row pitch in floats (pad: conflict-free gathers)
#define MT      (NN / TM)     // 128 m-tiles
#define NT      (NN / NC)     // 8  n-chunks
#define NWG     (MT * NT)     // 1024 main workgroups

typedef float v4f __attribute__((ext_vector_type(4)));

// Low 32 bits of a generic pointer to __shared__ == LDS byte offset within the
// wave's allocation; exactly what the async
// copy's VDST wants.
__device__ __forceinline__ uint32_t lds_off(const void* p) {
    return (uint32_t)(uintptr_t)p;
}

// CDNA5 async data mover: 16B global -> LDS, tracked by ASYNCcnt.
__device__ __forceinline__ void async_g2l_b128(uint32_t lds_byte, uint32_t voff_byte,
                                               const void* sbase) {
    asm volatile("global_load_async_to_lds_b128 %0, %1, %2"
                 :: "v"(lds_byte), "v"(voff_byte), "s"(sbase)
                 : "memory");
}

__device__ __forceinline__ void wait_async0() {
    asm volatile("s_wait_asynccnt 0" ::: "memory");
}

__global__ void zero_f32_kernel(float* __restrict__ p, int nfloats) {
    int i = blockIdx.x * blockDim.x + threadIdx.x;
    if (i < nfloats) p[i] = 0.0f;
}

// Main streaming kernel: one pass over rho; S/D staged via async copies.
__global__ __launch_bounds__(256)
void ObservableCalculator_main_kernel(const float* __restrict__ rho,
                                      const float* __restrict__ S,
                                      const float* __restrict__ D,
                                      float* __restrict__ diag_ws,   // [16][2048]
                                      float* __restrict__ dip_ws) {  // [NWG][24]
    __shared__ float lds[4 * NC * PITCH + 24];   // S,D0,D1,D2 tiles + dipole accum
    float* dipacc = &lds[4 * NC * PITCH];

    const int tid  = threadIdx.x;
    const int m0   = blockIdx.x * TM;                    // m-tile
    const int n0   = blockIdx.y * NC;                    // n-chunk
    const int wgid = blockIdx.y * gridDim.x + blockIdx.x;

    if (tid < 24) dipacc[tid] = 0.0f;

    // ---- Stage S[n0:n0+NC, m0:m0+16] and D_c tiles into LDS (async DMA) ----
    const uint32_t lds0 = lds_off(&lds[0]);
    #pragma unroll
    for (int mat = 0; mat < 4; ++mat) {
        const float* gbase = (mat == 0) ? S : (D + (size_t)(mat - 1) * NN * NN);
        const uint32_t lbase = (uint32_t)mat * NC * PITCH;
        #pragma unroll
        for (int q = 0; q < 4; ++q) {
            const int L  = q * 256 + tid;     // 1024 x b128 per matrix
            const int r  = L >> 2;            // tile row (n-local), 0..255
            const int c4 = (L & 3) * 4;       // 4-float column group, 0..12
            const uint32_t voff  = ((uint32_t)(n0 + r) * NN + (uint32_t)(m0 + c4)) * 4u;
            const uint32_t laddr = lds0 + (lbase + (uint32_t)r * PITCH + (uint32_t)c4) * 4u;
            async_g2l_b128(laddr, voff, gbase);
        }
    }
    wait_async0();
    __syncthreads();

    // ---- Gather this thread's transposed S/D values into registers (once) ----
    const int j = tid >> 4;       // my row within tile: m = m0 + j
    const int t = tid & 15;       // my n-lane group
    float sreg[16], dreg0[16], dreg1[16], dreg2[16];
    #pragma unroll
    for (int kk = 0; kk < 4; ++kk) {
        #pragma unroll
        for (int i = 0; i < 4; ++i) {
            const int nl  = kk * 64 + 4 * t + i;       // n-local index
            const int idx = nl * PITCH + j;
            sreg [kk * 4 + i] = lds[idx];
            dreg0[kk * 4 + i] = lds[1 * NC * PITCH + idx];
            dreg1[kk * 4 + i] = lds[2 * NC * PITCH + idx];
            dreg2[kk * 4 + i] = lds[3 * NC * PITCH + idx];
        }
    }

    // ---- Stream rho once across all 16 (b,s) planes ----
    const int m = m0 + j;
    const uint64_t rowbase = (uint64_t)m * NN + (uint64_t)(n0 + 4 * t);
    float dip0 = 0.0f, dip1 = 0.0f, dip2 = 0.0f;

    for (int b = 0; b < 8; ++b) {
        #pragma unroll
        for (int s = 0; s < 2; ++s) {
            const int plane = b * 2 + s;
            const v4f* rp = (const v4f*)(rho + (uint64_t)plane * NN * NN + rowbase);
            float diagacc = 0.0f;
            #pragma unroll
            for (int kk = 0; kk < 4; ++kk) {
                const v4f r = __builtin_nontemporal_load(rp + kk * 16); // +64 floats
                diagacc += r.x * sreg [4*kk+0] + r.y * sreg [4*kk+1]
                         + r.z * sreg [4*kk+2] + r.w * sreg [4*kk+3];
                dip0    += r.x * dreg0[4*kk+0] + r.y * dreg0[4*kk+1]
                         + r.z * dreg0[4*kk+2] + r.w * dreg0[4*kk+3];
                dip1    += r.x * dreg1[4*kk+0] + r.y * dreg1[4*kk+1]
                         + r.z * dreg1[4*kk+2] + r.w * dreg1[4*kk+3];
                dip2    += r.x * dreg2[4*kk+0] + r.y * dreg2[4*kk+1]
                         + r.z * dreg2[4*kk+2] + r.w * dreg2[4*kk+3];
            }
            // reduce diag over the 16 n-lanes of this row (xor stays in 16-group)
            diagacc += __shfl_xor(diagacc, 1);
            diagacc += __shfl_xor(diagacc, 2);
            diagacc += __shfl_xor(diagacc, 4);
            diagacc += __shfl_xor(diagacc, 8);
            if (t == 0) atomicAdd(&diag_ws[plane * NN + m], diagacc);
        }
        // dipole: sum spins done implicitly; reduce over full wave, bank into LDS
        #pragma unroll
        for (int msk = 1; msk < 32; msk <<= 1) {
            dip0 += __shfl_xor(dip0, msk);
            dip1 += __shfl_xor(dip1, msk);
            dip2 += __shfl_xor(dip2, msk);
        }
        if ((tid & 31) == 0) {
            atomicAdd(&dipacc[b * 3 + 0], dip0);
            atomicAdd(&dipacc[b * 3 + 1], dip1);
            atomicAdd(&dipacc[b * 3 + 2], dip2);
        }
        dip0 = dip1 = dip2 = 0.0f;
    }
    __syncthreads();
    if (tid < 24) dip_ws[wgid * 24 + tid] = dipacc[tid];
}

// populations[plane, a] = sum_{m: map[m]==a} diag[plane, m]
__global__ void pop_scatter_kernel(const float* __restrict__ diag_ws,
                                   const int* __restrict__ map,
                                   float* __restrict__ out) {
    const int plane = blockIdx.x;           // 0..15
    __shared__ float pop[NATOMS];
    pop[threadIdx.x] = 0.0f;
    __syncthreads();
    for (int m = threadIdx.x; m < NN; m += 256)
        atomicAdd(&pop[map[m]], diag_ws[plane * NN + m]);
    __syncthreads();
    out[plane * NATOMS + threadIdx.x] = pop[threadIdx.x];
}

// dipole[o] = sum over the 1024 per-WG partials
__global__ void dip_reduce_kernel(const float* __restrict__ dip_ws,
                                  float* __restrict__ out) {
    const int o = blockIdx.x;               // 0..23
    float s = 0.0f;
    for (int k = threadIdx.x; k < NWG; k += 256) s += dip_ws[k * 24 + o];
    #pragma unroll
    for (int msk = 1; msk < 32; msk <<= 1) s += __shfl_xor(s, msk);
    __shared__ float red[8];
    if ((threadIdx.x & 31) == 0) red[threadIdx.x >> 5] = s;
    __syncthreads();
    if (threadIdx.x == 0) {
        float tot = 0.0f;
        #pragma unroll
        for (int w = 0; w < 8; ++w) tot += red[w];
        out[NPLANE * NATOMS + o] = tot;     // dipole after 4096 population floats
    }
}

extern "C" void kernel_launch(void* const* d_in, const int* in_sizes, int n_in,
                              void* d_out, int out_size, void* d_ws, size_t ws_size,
                              hipStream_t stream) {
    const float* rho = (const float*)d_in[0];   // (8,2,2048,2048)
    const float* S   = (const float*)d_in[1];   // (2048,2048)
    const float* D   = (const float*)d_in[2];   // (3,2048,2048)
    const int*   map = (const int*)d_in[3];     // (2048,)
    float* out = (float*)d_out;                 // 4096 pop + 24 dipole

    float* diag_ws = (float*)d_ws;              // 16*2048 floats
    float* dip_ws  = diag_ws + NPLANE * NN;     // NWG*24 floats (fully overwritten)

    zero_f32_kernel<<<(NPLANE * NN + 255) / 256, 256, 0, stream>>>(diag_ws, NPLANE * NN);

    dim3 grid(MT, NT);                          // 128 x 8 = 1024 workgroups
    ObservableCalculator_main_kernel<<<grid, 256, 0, stream>>>(rho, S, D, diag_ws, dip_ws);

    pop_scatter_kernel<<<NPLANE, 256, 0, stream>>>(diag_ws, map, out);
    dip_reduce_kernel<<<24, 256, 0, stream>>>(dip_ws, out);
}